// Net_new_23364622090710
// MI455X (gfx1250) — compile-verified
//
#include <hip/hip_runtime.h>
#include <math.h>

// ---------------- problem sizes ----------------
#define S    512      // batch
#define D    68       // feature dim
#define DPD  80       // D padded to 16
#define H1   128      // hidden 1
#define H2   64       // hidden 2
#define DD   4624     // 68*68  (289*16)
#define KO   4352     // 64*68  (outer-product K)
#define KOB  4420     // KO + D (b3 tail rows), mult of 4
#define NPTS 4        // power-flow iterations

typedef __attribute__((ext_vector_type(2))) float v2f;
typedef __attribute__((ext_vector_type(8))) float v8f;

// ---------------- generic f32 WMMA GEMM ----------------
// C[M,N] = epilogue( A[M,K] @ B[K,N] ), wave32, one 16x16 tile per wave,
// K-loop in steps of 4 using V_WMMA_F32_16X16X4_F32; flags are compile-time.
struct GemmP {
  const float* A;   int sA;   int lda;
  const float* A2;  int sA2;  int lda2; int k_split;  // outer-product operands
  const float* Asc; int sAsc;                         // sigmoid' row scale
  const float* B;   int sB;   int ldb;
  float*       C;   int sC;   int ldc;
  const float* bias;
  const float* Add; int sAdd;                         // scatter-add (stride-68) source
  int M, N, K;
};

// AMODE: 0 plain, 2 outer-product A[m,q]*A2[m,i] (+ b3 tail rows = A2 direct)
template<int AMODE, int ATRANS, int ASCALE>
__device__ __forceinline__ v2f ldA2(const GemmP& p, const float* A, const float* A2,
                                    const float* Asc, int m, int ka) {
  v2f r;
  if constexpr (AMODE == 2) {
    if (ka < p.k_split) {
      const int q = ka / D, i = ka - q * D;              // ka even => pair never crosses a q boundary
      const v2f u = *(const v2f*)(A2 + m * p.lda2 + i);
      const float bq = A[m * p.lda + q];
      r.x = bq * u.x; r.y = bq * u.y;
    } else {
      r = *(const v2f*)(A2 + m * p.lda2 + (ka - p.k_split));
    }
  } else if constexpr (ATRANS) {
    r.x = A[ka * p.lda + m];
    r.y = A[(ka + 1) * p.lda + m];
  } else {
    r = *(const v2f*)(A + m * p.lda + ka);               // contiguous K-pair -> b64 load
  }
  if constexpr (ASCALE) {
    const v2f s = *(const v2f*)(Asc + ka);
    r.x *= s.x * (1.f - s.x);
    r.y *= s.y * (1.f - s.y);
  }
  return r;
}

template<int BTRANS>
__device__ __forceinline__ v2f ldB2(const GemmP& p, const float* B, int ka, int n) {
  v2f r;
  if constexpr (BTRANS) {
    r = *(const v2f*)(B + n * p.ldb + ka);               // contiguous K-pair -> b64 load
  } else {
    r.x = B[ka * p.ldb + n];
    r.y = B[(ka + 1) * p.ldb + n];
  }
  return r;
}

template<int AMODE, int ATRANS, int ASCALE, int BTRANS, int BIAS, int ACT, int ADDM>
__global__ void __launch_bounds__(32) gemm_wmma(GemmP p) {
  const int z = blockIdx.z;
  const float* A   = p.A   + (size_t)z * p.sA;
  const float* A2  = p.A2  + (size_t)z * p.sA2;
  const float* Asc = p.Asc + (size_t)z * p.sAsc;
  const float* B   = p.B   + (size_t)z * p.sB;
  float*       C   = p.C   + (size_t)z * p.sC;

  const int m0   = blockIdx.x * 16, n0 = blockIdx.y * 16;
  const int lane = threadIdx.x;          // 0..31 (wave32)
  const int half = lane >> 4;            // K sub-pair selector per ISA f32 A/B layout
  const int l    = lane & 15;
  const int m    = m0 + l;
  const int n    = n0 + l;

  v8f acc = {0.f,0.f,0.f,0.f,0.f,0.f,0.f,0.f};

  auto step = [&](int kb) {
    const int ka = kb + 2 * half;        // lanes 0-15: K=kb,kb+1 ; lanes 16-31: K=kb+2,kb+3
    const v2f av = ldA2<AMODE, ATRANS, ASCALE>(p, A, A2, Asc, m, ka);
    const v2f bv = ldB2<BTRANS>(p, B, ka, n);
    acc = __builtin_amdgcn_wmma_f32_16x16x4_f32(false, av, false, bv, (short)0, acc, false, false);
  };

  int k0 = 0;
  for (; k0 + 16 <= p.K; k0 += 16) {     // software-pipelined main loop: 4 WMMA / iter
    // pull next panels toward the wave while WMMAs execute (speculative, fault-safe)
    if constexpr (AMODE == 0 && !ATRANS) __builtin_prefetch(A + m * p.lda + k0 + 32, 0, 1);
    if constexpr (BTRANS)                __builtin_prefetch(B + n * p.ldb + k0 + 32, 0, 1);
    else                                 __builtin_prefetch(B + (k0 + 32) * p.ldb + n, 0, 1);
#pragma unroll
    for (int kk = 0; kk < 16; kk += 4) step(k0 + kk);
  }
  for (; k0 < p.K; k0 += 4) step(k0);    // tail (K % 16)

#pragma unroll
  for (int r = 0; r < 8; ++r) {          // D layout: lanes<16 -> M=r, lanes>=16 -> M=r+8
    const int mm = m0 + r + 8 * half;
    float v = acc[r];
    if constexpr (BIAS) v += p.bias[n];
    if constexpr (ACT)  v = 1.f / (1.f + __expf(-v));
    if constexpr (ADDM) { if (mm < D && n < D) v += (p.Add + (size_t)z * p.sAdd)[mm * D + n]; }
    C[mm * p.ldc + n] = v;
  }
}

// ---------------- small helper kernels ----------------
__global__ void pad_x_kernel(const float* x, float* z0) {
  int idx = blockIdx.x * 256 + threadIdx.x; if (idx >= S * DPD) return;
  int s = idx / DPD, c = idx % DPD;
  z0[idx] = (c < D) ? x[s * D + c] : 0.f;
}
__global__ void pad_w1_kernel(const float* W1, float* W1p) {
  int idx = blockIdx.x * 256 + threadIdx.x; if (idx >= DPD * H1) return;
  int r = idx / H1, c = idx % H1;
  W1p[idx] = (r < D) ? W1[r * H1 + c] : 0.f;
}
__global__ void build_w3aug_kernel(const float* W3, const float* b3, float* W3aug) {
  int idx = blockIdx.x * 256 + threadIdx.x; if (idx >= KOB * DPD) return;
  int r = idx / DPD, j = idx % DPD;
  float v = 0.f;
  if (j < D) {
    if (r < KO) { int q = r / D, i = r - q * D; v = W3[q * DD + i * D + j]; }
    else        { int i = r - KO; v = b3[i * D + j]; }
  }
  W3aug[idx] = v;
}
__global__ void build_w3perm_kernel(const float* W3, float* W3pm) {
  int idx = blockIdx.x * 256 + threadIdx.x; if (idx >= KO * DPD) return;
  int r = idx / DPD, m = idx % DPD;
  int q = r / D, j = r - q * D;
  W3pm[idx] = (m < D) ? W3[q * DD + m * D + j] : 0.f;
}
__global__ void db_kernel(const float* E, const float* B1, float* Db) {
  int idx = blockIdx.x * 256 + threadIdx.x; if (idx >= S * DPD * H2) return;
  int s = idx / (DPD * H2); int q = idx % H2;
  float b = B1[s * H2 + q];
  Db[idx] = E[idx] * b * (1.f - b);
}
__global__ void ptflat_kernel(const float* P, float* PT) {   // PT[s,k*68+j] = P[s][j,k]
  int idx = blockIdx.x * 256 + threadIdx.x; if (idx >= S * DD) return;
  int s = idx / DD, c = idx % DD; int k = c / D, j = c - k * D;
  PT[idx] = P[s * DPD * DPD + j * DPD + k];
}
__global__ void pdtflat_kernel(const float* PDT, float* Pf) { // Pf[s,q*68+j] = PDT[s][q,j]
  int idx = blockIdx.x * 256 + threadIdx.x; if (idx >= S * KO) return;
  int s = idx / KO, c = idx % KO; int q = c / D, j = c - q * D;
  Pf[idx] = PDT[s * (H2 * DPD) + q * DPD + j];
}

// per-sample Gauss-Jordan inverse of the 68x68 block of Kmat, partial pivoting, in LDS
__global__ void __launch_bounds__(256) invert_kernel(const float* Kall, float* Pall) {
  const int s = blockIdx.x, t = threadIdx.x;
  __shared__ float Ms[D][72];
  __shared__ float Is[D][72];
  __shared__ float red[128];
  __shared__ int   redi[128];
  for (int idx = t; idx < D * D; idx += 256) {
    int r = idx / D, c = idx - r * D;
    Ms[r][c] = Kall[s * DPD * DPD + r * DPD + c];
    Is[r][c] = (r == c) ? 1.f : 0.f;
  }
  __syncthreads();
  for (int k = 0; k < D; ++k) {
    if (t < 128) {
      float v = -1.f; int ri = k;
      if (t < D && t >= k) { v = fabsf(Ms[t][k]); ri = t; }
      red[t] = v; redi[t] = ri;
    }
    __syncthreads();
    for (int off = 64; off > 0; off >>= 1) {
      if (t < off) { if (red[t + off] > red[t]) { red[t] = red[t + off]; redi[t] = redi[t + off]; } }
      __syncthreads();
    }
    const int pr = redi[0];
    if (pr != k) {
      if (t < D) { float tmp = Ms[k][t]; Ms[k][t] = Ms[pr][t]; Ms[pr][t] = tmp; }
      else if (t >= 128 && t < 128 + D) { int c = t - 128; float tmp = Is[k][c]; Is[k][c] = Is[pr][c]; Is[pr][c] = tmp; }
    }
    __syncthreads();
    const float inv = 1.f / Ms[k][k];
    if (t < D) Ms[k][t] *= inv;
    else if (t >= 128 && t < 128 + D) Is[k][t - 128] *= inv;
    __syncthreads();
    if (t < D) red[t] = Ms[t][k];
    __syncthreads();
    for (int idx = t; idx < D * 2 * D; idx += 256) {
      int r = idx / (2 * D), c = idx - r * (2 * D);
      if (r != k) {
        float f = red[r];
        if (c < D) Ms[r][c] -= f * Ms[k][c];
        else       Is[r][c - D] -= f * Is[k][c - D];
      }
    }
    __syncthreads();
  }
  for (int idx = t; idx < DPD * DPD; idx += 256) {
    int r = idx / DPD, c = idx - r * DPD;
    Pall[s * DPD * DPD + idx] = (r < D && c < D) ? Is[r][c] : 0.f;
  }
}

// g = -(Db@(w3P + (1-2b)e) + W1@(s1(1-2a)f) + T3) per sample
__global__ void __launch_bounds__(256) g_assemble(const float* A1, const float* B1, const float* E,
                                                  const float* CPt, const float* F, const float* Db,
                                                  const float* w3Pb, const float* T3b,
                                                  const float* W1, const float* W2, float* gvec) {
  const int s = blockIdx.x, t = threadIdx.x;
  __shared__ float v1[H2];
  __shared__ float v2[H1];
  if (t < H2) {
    float b = B1[s * H2 + t];
    float e = 0.f;
    for (int k = 0; k < D; ++k) e += E[s * DPD * H2 + k * H2 + t] * CPt[s * H2 * DPD + t * DPD + k];
    v1[t] = w3Pb[s * H2 + t] + (1.f - 2.f * b) * e;
  }
  if (t >= 128 && t < 128 + H1) {
    int p = t - 128;
    float a = A1[s * H1 + p];
    float f = 0.f;
    for (int q = 0; q < H2; ++q) {
      float b = B1[s * H2 + q];
      f += W2[p * H2 + q] * b * (1.f - b) * F[s * H1 * H2 + p * H2 + q];
    }
    v2[p] = a * (1.f - a) * (1.f - 2.f * a) * f;
  }
  __syncthreads();
  if (t < DPD) {
    float gm = 0.f;
    if (t < D) {
      float acc = T3b[s * DPD + t];
      for (int q = 0; q < H2; ++q) acc += Db[s * DPD * H2 + t * H2 + q] * v1[q];
      for (int p = 0; p < H1; ++p) acc += W1[t * H1 + p] * v2[p];
      gm = -acc;
    }
    gvec[s * DPD + t] = gm;
  }
}

// a_new = (use_g? g : 0) + (use_K? K @ a_old : 0)
__global__ void __launch_bounds__(128) chain_kernel(const float* Kall, const float* g,
                                                    const float* a_old, float* a_new,
                                                    int use_g, int use_K) {
  const int s = blockIdx.x, t = threadIdx.x;
  __shared__ float av[DPD];
  if (t < DPD) av[t] = use_K ? a_old[s * DPD + t] : 0.f;
  __syncthreads();
  if (t < DPD) {
    float acc = use_g ? g[s * DPD + t] : 0.f;
    if (use_K && t < D) for (int j = 0; j < D; ++j) acc += Kall[s * DPD * DPD + t * DPD + j] * av[j];
    if (t >= D) acc = 0.f;
    a_new[s * DPD + t] = acc;
  }
}

__global__ void __launch_bounds__(128) final_kernel(const float* z4, const float* avec,
                                                    const float* Wx, const float* bx, float* out) {
  const int s = blockIdx.x, t = threadIdx.x;
  __shared__ float red[128];
  float part = 0.f;
  if (t < D) part = z4[s * DPD + t] * Wx[t];
  red[t] = part; __syncthreads();
  for (int off = 64; off > 0; off >>= 1) { if (t < off) red[t] += red[t + off]; __syncthreads(); }
  if (t == 0) out[s * 69] = 1.f / (1.f + __expf(-(red[0] + bx[0])));
  if (t < D) out[s * 69 + 1 + t] = avec[s * DPD + t];
}

// ---------------- host orchestration ----------------
template<int AMODE, int ATRANS, int ASCALE, int BTRANS, int BIAS, int ACT, int ADDM>
static void launch_gemm(const GemmP& p, int zdim, hipStream_t stream) {
  dim3 g((unsigned)(p.M / 16), (unsigned)(p.N / 16), (unsigned)zdim);
  gemm_wmma<AMODE, ATRANS, ASCALE, BTRANS, BIAS, ACT, ADDM><<<g, dim3(32), 0, stream>>>(p);
}

extern "C" void kernel_launch(void* const* d_in, const int* in_sizes, int n_in,
                              void* d_out, int out_size, void* d_ws, size_t ws_size,
                              hipStream_t stream) {
  (void)in_sizes; (void)n_in; (void)out_size; (void)ws_size; // needs ~128 MB workspace
  const float* x  = (const float*)d_in[0];
  const float* W1 = (const float*)d_in[1];
  const float* b1 = (const float*)d_in[2];
  const float* W2 = (const float*)d_in[3];
  const float* b2 = (const float*)d_in[4];
  const float* W3 = (const float*)d_in[5];
  const float* b3 = (const float*)d_in[6];
  const float* Wx = (const float*)d_in[7];
  const float* bx = (const float*)d_in[8];
  float* out = (float*)d_out;

  float* ws = (float*)d_ws; size_t off = 0;
  auto alloc = [&](size_t n) { float* p = ws + off; off += n; return p; };
  float* zpts = alloc((size_t)(NPTS + 1) * S * DPD);
  float* A1   = alloc((size_t)S * H1);
  float* B1   = alloc((size_t)S * H2);
  float* Ebuf = alloc((size_t)S * DPD * H2);
  float* Dbuf = alloc((size_t)S * DPD * H2);
  float* Cbuf = alloc((size_t)S * H2 * DPD);
  float* Hall = alloc((size_t)S * DD);
  float* Kmat = alloc((size_t)S * DPD * DPD);
  float* Pmat = alloc((size_t)S * DPD * DPD);
  float* CPtb = alloc((size_t)S * H2 * DPD);
  float* PDTb = alloc((size_t)S * H2 * DPD);
  float* Fbuf = alloc((size_t)S * H1 * H2);
  float* PTf  = alloc((size_t)S * DD);
  float* PDTf = alloc((size_t)S * KO);
  float* w3Pb = alloc((size_t)S * H2);
  float* T3b  = alloc((size_t)S * DPD);
  float* gvec = alloc((size_t)S * DPD);
  float* avA  = alloc((size_t)S * DPD);
  float* avB  = alloc((size_t)S * DPD);
  float* W1p  = alloc((size_t)DPD * H1);
  float* W3aug= alloc((size_t)KOB * DPD);
  float* W3pm = alloc((size_t)KO * DPD);

  auto g1 = [](int n) { return dim3((unsigned)((n + 255) / 256)); };

  // padded / permuted weights (redone every call: deterministic)
  pad_x_kernel<<<g1(S * DPD), 256, 0, stream>>>(x, zpts);
  pad_w1_kernel<<<g1(DPD * H1), 256, 0, stream>>>(W1, W1p);
  build_w3aug_kernel<<<g1(KOB * DPD), 256, 0, stream>>>(W3, b3, W3aug);
  build_w3perm_kernel<<<g1(KO * DPD), 256, 0, stream>>>(W3, W3pm);

  auto hnet12 = [&](const float* zc) {
    GemmP p = {}; p.A = zc; p.lda = DPD; p.B = W1p; p.ldb = H1; p.C = A1; p.ldc = H1;
    p.M = S; p.N = H1; p.K = DPD; p.bias = b1;
    launch_gemm<0,0,0,0,1,1,0>(p, 1, stream);
    GemmP q = {}; q.A = A1; q.lda = H1; q.B = W2; q.ldb = H2; q.C = B1; q.ldc = H2;
    q.M = S; q.N = H2; q.K = H1; q.bias = b2;
    launch_gemm<0,0,0,0,1,1,0>(q, 1, stream);
  };

  // ---- forward chain: z_{i+1} = (b x u) @ W3 + u @ b3r, via outer-product GEMM ----
  for (int it = 0; it < NPTS; ++it) {
    const float* zc = zpts + (size_t)it * S * DPD;
    float* zn = zpts + (size_t)(it + 1) * S * DPD;
    hnet12(zc);
    GemmP p = {}; p.A = B1; p.lda = H2; p.A2 = zc; p.lda2 = DPD; p.k_split = KO;
    p.B = W3aug; p.ldb = DPD; p.C = zn; p.ldc = DPD; p.M = S; p.N = DPD; p.K = KOB;
    launch_gemm<2,0,0,0,0,0,0>(p, 1, stream);
  }

  // ---- backward: points z4..z0, carrying adjoint a ----
  float* aCur = avA; float* aNext = avB;
  for (int pt = NPTS; pt >= 0; --pt) {
    const float* zc = zpts + (size_t)pt * S * DPD;
    hnet12(zc);
    { // E[k,q] = sum_p (W1[k,p] s1[p]) W2[p,q], per sample
      GemmP p = {}; p.A = W1p; p.lda = H1; p.Asc = A1; p.sAsc = H1;
      p.B = W2; p.ldb = H2; p.C = Ebuf; p.sC = DPD * H2; p.ldc = H2;
      p.M = DPD; p.N = H2; p.K = H1;
      launch_gemm<0,0,1,0,0,0,0>(p, S, stream);
    }
    db_kernel<<<g1(S * DPD * H2), 256, 0, stream>>>(Ebuf, B1, Dbuf);
    { // C[s,q,j] = sum_i z_i W3r[q,i,j]  (batch over q)
      GemmP p = {}; p.A = zc; p.lda = DPD; p.B = W3aug; p.sB = D * DPD; p.ldb = DPD;
      p.C = Cbuf; p.sC = DPD; p.ldc = H2 * DPD; p.M = S; p.N = DPD; p.K = D;
      launch_gemm<0,0,0,0,0,0,0>(p, H2, stream);
    }
    { // Hall = B1 @ W3 + b3
      GemmP p = {}; p.A = B1; p.lda = H2; p.B = W3; p.ldb = DD; p.C = Hall; p.ldc = DD;
      p.M = S; p.N = DD; p.K = H2; p.bias = b3;
      launch_gemm<0,0,0,0,1,0,0>(p, 1, stream);
    }
    { // K = Db @ C + H, per sample
      GemmP p = {}; p.A = Dbuf; p.sA = DPD * H2; p.lda = H2; p.B = Cbuf; p.sB = H2 * DPD; p.ldb = DPD;
      p.C = Kmat; p.sC = DPD * DPD; p.ldc = DPD; p.M = DPD; p.N = DPD; p.K = H2;
      p.Add = Hall; p.sAdd = DD;
      launch_gemm<0,0,0,0,0,0,1>(p, S, stream);
    }
    if (pt > 0) {
      invert_kernel<<<S, 256, 0, stream>>>(Kmat, Pmat);
      { // CPt[q,k] = sum_j C[q,j] P[j,k]
        GemmP p = {}; p.A = Cbuf; p.sA = H2 * DPD; p.lda = DPD; p.B = Pmat; p.sB = DPD * DPD; p.ldb = DPD;
        p.C = CPtb; p.sC = H2 * DPD; p.ldc = DPD; p.M = H2; p.N = DPD; p.K = DPD;
        launch_gemm<0,0,0,0,0,0,0>(p, S, stream);
      }
      { // PDT[q,j] = sum_k Db[k,q] P[j,k]
        GemmP p = {}; p.A = Dbuf; p.sA = DPD * H2; p.lda = H2;
        p.B = Pmat; p.sB = DPD * DPD; p.ldb = DPD;
        p.C = PDTb; p.sC = H2 * DPD; p.ldc = DPD; p.M = H2; p.N = DPD; p.K = DPD;
        launch_gemm<0,1,0,1,0,0,0>(p, S, stream);
      }
      { // F[p,q] = sum_k W1[k,p] CPt[q,k]
        GemmP p = {}; p.A = W1p; p.lda = H1;
        p.B = CPtb; p.sB = H2 * DPD; p.ldb = DPD;
        p.C = Fbuf; p.sC = H1 * H2; p.ldc = H2; p.M = H1; p.N = H2; p.K = DPD;
        launch_gemm<0,1,0,1,0,0,0>(p, S, stream);
      }
      ptflat_kernel<<<g1(S * DD), 256, 0, stream>>>(Pmat, PTf);
      pdtflat_kernel<<<g1(S * KO), 256, 0, stream>>>(PDTb, PDTf);
      { // w3P[s,q] = sum_c PT[s,c] W3[q,c]
        GemmP p = {}; p.A = PTf; p.lda = DD; p.B = W3; p.ldb = DD;
        p.C = w3Pb; p.ldc = H2; p.M = S; p.N = H2; p.K = DD;
        launch_gemm<0,0,0,1,0,0,0>(p, 1, stream);
      }
      { // T3[s,m] = sum_c PDTf[s,c] W3perm[c,m]
        GemmP p = {}; p.A = PDTf; p.lda = KO; p.B = W3pm; p.ldb = DPD;
        p.C = T3b; p.ldc = DPD; p.M = S; p.N = DPD; p.K = KO;
        launch_gemm<0,0,0,0,0,0,0>(p, 1, stream);
      }
      g_assemble<<<S, 256, 0, stream>>>(A1, B1, Ebuf, CPtb, Fbuf, Dbuf, w3Pb, T3b, W1, W2, gvec);
    }
    const int use_g = (pt > 0), use_K = (pt < NPTS);
    chain_kernel<<<S, 128, 0, stream>>>(Kmat, gvec, aCur, aNext, use_g, use_K);
    float* t = aCur; aCur = aNext; aNext = t;
  }

  final_kernel<<<S, 128, 0, stream>>>(zpts + (size_t)NPTS * S * DPD, aCur, Wx, bx, out);
}